// DiffFDN_26190710571641
// MI455X (gfx1250) — compile-verified
//
#include <hip/hip_runtime.h>
#include <math.h>

#define NM 16
#define FBINS 192001
#define NFFT_ 384000
#define NN1 480
#define NN2 800
#define CH 80              // K-chunk staged per TDM transfer (divides 480 and 800)
#define PI_D 3.14159265358979323846

static_assert(NN1 * NN2 == NFFT_, "factorization");
static_assert(NN1 % CH == 0 && NN2 % CH == 0, "chunking");

typedef __attribute__((ext_vector_type(2))) float v2f;
typedef __attribute__((ext_vector_type(8))) float v8f;
typedef __attribute__((ext_vector_type(4))) unsigned int v4u;
typedef __attribute__((ext_vector_type(4))) int v4i;
typedef __attribute__((ext_vector_type(8))) int v8i;

// ---------------- workspace layout (float offsets) ----------------
constexpr size_t OFF_AG    = 0;                                   // 256
constexpr size_t OFF_GAMMA = 256;                                 // 16
constexpr size_t OFF_M     = 272;                                 // 16
constexpr size_t OFF_MAXP  = 288;                                 // 256 partials + 1 final
constexpr size_t OFF_HS    = 1024;                                // 2*FBINS
constexpr size_t OFF_XRE   = OFF_HS + 2 * (size_t)FBINS + 2;      // 384000
constexpr size_t OFF_XIM   = OFF_XRE + (size_t)NN1 * NN2;
constexpr size_t OFF_W2RE  = OFF_XIM + (size_t)NN1 * NN2;         // 640000
constexpr size_t OFF_W2IM  = OFF_W2RE + (size_t)NN2 * NN2;
constexpr size_t OFF_W1RE  = OFF_W2IM + (size_t)NN2 * NN2;        // 230400
constexpr size_t OFF_W1IM  = OFF_W1RE + (size_t)NN1 * NN1;
constexpr size_t OFF_ZRE   = OFF_W1IM + (size_t)NN1 * NN1;        // 384000
constexpr size_t OFF_ZIM   = OFF_ZRE + (size_t)NN1 * NN2;
constexpr size_t OFF_HU    = OFF_ZIM + (size_t)NN1 * NN2;         // 384000

__constant__ float c_delays[NM] = {809.f, 877.f, 937.f, 1049.f, 1151.f, 1249.f, 1373.f, 1499.f,
                                   1617.f, 1753.f, 1879.f, 2003.f, 2131.f, 2269.f, 2393.f, 2521.f};

// ---------------- Stage 0: A = expm(skew(X)); AG = A*diag(gamma); m, gamma ----------------
__global__ void prep_kernel(const float* __restrict__ X, float* __restrict__ ws) {
    __shared__ float sT[256], sP[256], sG[16];
    __shared__ int sSq;
    const int tid = threadIdx.x;
    const int r = tid >> 4, c = tid & 15;

    float s = (c > r) ? X[r * 16 + c] : ((c < r) ? -X[c * 16 + r] : 0.0f);
    sT[tid] = s;
    __syncthreads();

    if (tid == 0) {
        float nrm = 0.0f;
        for (int i = 0; i < 16; ++i) {
            float rs = 0.0f;
            for (int j = 0; j < 16; ++j) rs += fabsf(sT[i * 16 + j]);
            nrm = fmaxf(nrm, rs);
        }
        int sq = 0;
        while (nrm > 0.25f && sq < 32) { nrm *= 0.5f; ++sq; }
        sSq = sq;
    }
    __syncthreads();
    const int sq = sSq;
    sT[tid] = s * ldexpf(1.0f, -sq);       // T = S / 2^sq
    sP[tid] = (r == c) ? 1.0f : 0.0f;
    __syncthreads();

    for (int it = 18; it >= 1; --it) {     // degree-18 Taylor via Horner
        float acc = 0.0f;
#pragma unroll
        for (int j = 0; j < 16; ++j) acc += sT[r * 16 + j] * sP[j * 16 + c];
        __syncthreads();
        sP[tid] = ((r == c) ? 1.0f : 0.0f) + acc / (float)it;
        __syncthreads();
    }
    for (int q = 0; q < sq; ++q) {         // repeated squaring
        float acc = 0.0f;
#pragma unroll
        for (int j = 0; j < 16; ++j) acc += sP[r * 16 + j] * sP[j * 16 + c];
        __syncthreads();
        sP[tid] = acc;
        __syncthreads();
    }

    if (tid < 16) {
        float mean = 0.0f;
        for (int i = 0; i < 16; ++i) mean += c_delays[i];
        mean /= 16.0f;
        float var = 0.0f;
        for (int i = 0; i < 16; ++i) { float d = c_delays[i] - mean; var += d * d; }
        var /= 15.0f;
        const float sd = sqrtf(var);
        const float m = ((c_delays[tid] - mean) / sd) * sd + mean;
        const float g = powf(0.9998f, m);
        ws[OFF_M + tid] = m;
        ws[OFF_GAMMA + tid] = g;
        sG[tid] = g;
    }
    __syncthreads();
    ws[OFF_AG + tid] = sP[tid] * sG[c];    // AG = A @ diag(gamma)
}

// ---------------- Stage 1: per-frequency 16x16 complex solve (half-wave per bin) ----------------
__global__ void solve_kernel(const float* __restrict__ xre, const float* __restrict__ xim,
                             const float* __restrict__ Bv, const float* __restrict__ Cv,
                             const float* __restrict__ ws, float* __restrict__ outH,
                             float* __restrict__ hs) {
    const int tid = threadIdx.x;
    const int lane = tid & 31;
    const int i = lane & 15;
    const int base = lane & 16;
    int f = (int)((blockIdx.x * (unsigned)blockDim.x + tid) >> 4);
    if (f > FBINS - 1) f = FBINS - 1;

    const float* AG = ws + OFF_AG;
    const float mi = ws[OFF_M + i];

    const float cw = xre[f], sw = xim[f];
    const double w = atan2((double)sw, (double)cw);
    const double r2 = (double)cw * (double)cw + (double)sw * (double)sw;
    const double md = (double)mi;
    const double amp = exp(md * 0.5 * log(r2));
    double sph, cph;
    sincos(md * w, &sph, &cph);
    const float ddre = (float)(amp * cph);
    const float ddim = (float)(amp * sph);

    float Mre[16], Mim[16];
#pragma unroll
    for (int j = 0; j < 16; ++j) {
        Mre[j] = -AG[i * 16 + j];
        Mim[j] = 0.0f;
        if (j == i) { Mre[j] += ddre; Mim[j] += ddim; }
    }
    float rre = Bv[i], rim = 0.0f;

#pragma unroll
    for (int k = 0; k < 16; ++k) {
        const float pr = __shfl(Mre[k], base + k, 32);
        const float pi = __shfl(Mim[k], base + k, 32);
        const float inv = 1.0f / (pr * pr + pi * pi);
        const float fr = (Mre[k] * pr + Mim[k] * pi) * inv;
        const float fi = (Mim[k] * pr - Mre[k] * pi) * inv;
        const bool act = i > k;
#pragma unroll
        for (int j = k; j < 16; ++j) {
            const float ur = __shfl(Mre[j], base + k, 32);
            const float ui = __shfl(Mim[j], base + k, 32);
            if (act) {
                Mre[j] -= fr * ur - fi * ui;
                Mim[j] -= fr * ui + fi * ur;
            }
        }
        const float br = __shfl(rre, base + k, 32);
        const float bi = __shfl(rim, base + k, 32);
        if (act) {
            rre -= fr * br - fi * bi;
            rim -= fr * bi + fi * br;
        }
    }
#pragma unroll
    for (int k = 15; k >= 0; --k) {
        const float pr = __shfl(Mre[k], base + k, 32);
        const float pi = __shfl(Mim[k], base + k, 32);
        const float br = __shfl(rre, base + k, 32);
        const float bi = __shfl(rim, base + k, 32);
        const float inv = 1.0f / (pr * pr + pi * pi);
        const float xr = (br * pr + bi * pi) * inv;
        const float xi_ = (bi * pr - br * pi) * inv;
        if (i == k) { rre = xr; rim = xi_; }
        else if (i < k) {
            rre -= Mre[k] * xr - Mim[k] * xi_;
            rim -= Mre[k] * xi_ + Mim[k] * xr;
        }
    }

    const float cc = Cv[i];
    const float hre = rre * cc, him = rim * cc;
    outH[(size_t)f * 32 + 2 * i]     = hre;
    outH[(size_t)f * 32 + 2 * i + 1] = him;

    float sr = hre, si = him;
#pragma unroll
    for (int mm = 8; mm >= 1; mm >>= 1) {
        sr += __shfl_xor(sr, mm, 32);
        si += __shfl_xor(si, mm, 32);
    }
    const bool zero = (cw == 0.0f) && (sw == 0.0f);
    if (i == 0) {
        hs[2 * (size_t)f]     = zero ? 0.0f : sr;
        hs[2 * (size_t)f + 1] = zero ? 0.0f : si;
    }
}

// ---------------- DFT twiddle matrices ----------------
__global__ void w_kernel(float* __restrict__ wre, float* __restrict__ wim, int n) {
    const int idx = blockIdx.x * blockDim.x + threadIdx.x;
    if (idx >= n * n) return;
    const int a = idx / n, b = idx % n;
    const int p = (int)(((long long)a * b) % n);
    double s, c;
    sincos((double)p * (2.0 * PI_D / (double)n), &s, &c);
    wre[idx] = (float)c;
    wim[idx] = (float)s;
}

// ---------------- Hermitian extension: Xm[k1][k2] = x_full[k1+480*k2] ----------------
__global__ void pack_kernel(const float* __restrict__ hs, float* __restrict__ xr, float* __restrict__ xi) {
    const int idx = blockIdx.x * blockDim.x + threadIdx.x;
    if (idx >= NN1 * NN2) return;
    const int k1 = idx / NN2, k2 = idx % NN2;
    const int k = k1 + NN1 * k2;
    float re, im;
    if (k < FBINS) { re = hs[2 * (size_t)k]; im = hs[2 * (size_t)k + 1]; }
    else { const int kk = NFFT_ - k; re = hs[2 * (size_t)kk]; im = -hs[2 * (size_t)kk + 1]; }
    xr[idx] = re;
    xi[idx] = im;
}

// ---------------- TDM: DMA a [rows x 16] f32 tile (row stride = `stride` elems) into LDS ----------------
// D# packed per CDNA5 ISA 8.3/8.4: group0 {count=1, lds_addr, global_addr, type=2},
// group1 {data_size=4B, tensor_dim0=16, tensor_dim1=rows, tile_dim0=16, tile_dim1=rows,
//         tensor_dim0_stride=stride}; no cluster multicast, no padding, no iterate.
__device__ __forceinline__ void tdm_load_tile(unsigned ldsAddr, const float* gsrc,
                                              int rows, int stride) {
    const unsigned long long ga = (unsigned long long)(size_t)gsrc;
    v4u g0;
    g0.x = 1u;                                                   // count=1 valid descriptor
    g0.y = ldsAddr;                                              // LDS byte address
    g0.z = (unsigned)ga;                                         // global_addr[31:0]
    g0.w = (unsigned)((ga >> 32) & 0x01FFFFFFu) | 0x80000000u;   // global_addr[56:32] | type=2
    v8i g1;
    g1[0] = 0x20000;                                             // data_size=2 -> 4 bytes
    g1[1] = 16 << 16;                                            // tensor_dim0 = 16
    g1[2] = (rows & 0xFFFF) << 16;                               // tensor_dim1 = rows
    g1[3] = 16 << 16;                                            // tile_dim0 = 16
    g1[4] = rows & 0xFFFF;                                       // tile_dim1 = rows, tile_dim2 = 0
    g1[5] = stride;                                              // tensor_dim0_stride[31:0]
    g1[6] = 0;
    g1[7] = 0;
    const v4i z4 = {0, 0, 0, 0};
    const v8i z8 = {0, 0, 0, 0, 0, 0, 0, 0};
    // 6-arg form on this toolchain: (g0, g1, g2, g3, <extra group>, cpol)
    __builtin_amdgcn_tensor_load_to_lds(g0, g1, z4, z4, z8, 0);
}

// ---------------- complex GEMM: WMMA f32 16x16x4 + TDM double-buffered B panel ----------------
// Block = 8 waves = 128x16 output strip; all waves share one B panel per K-chunk.
// mode 0: apply twiddle e^{2pi i row*col/NFFT}, store complex; mode 1: store Re(C)/NFFT.
__global__ void gemm_cplx(const float* __restrict__ Are, const float* __restrict__ Aim,
                          const float* __restrict__ Bre, const float* __restrict__ Bim,
                          float* __restrict__ Cre, float* __restrict__ Cim,
                          int M, int K, int Nn, int mode) {
    __shared__ float sBre[2][CH * 16];
    __shared__ float sBim[2][CH * 16];
    const int tid = threadIdx.x;
    const int wave = tid >> 5, lane = tid & 31;
    const int colTiles = Nn >> 4;
    const int strip = blockIdx.x / colTiles;
    const int col0 = (blockIdx.x % colTiles) << 4;
    int row0 = strip * 128 + wave * 16;
    if (row0 > M - 16) row0 = M - 16;     // tail waves recompute last tile (identical writes)
    const int m = lane & 15;
    const int hi = lane >> 4;             // lanes 16..31 hold K+2/K+3 halves of fragments
    const int koff = hi * 2;
    const int nch = K / CH;

    v8f p1 = {}, p2 = {}, p3 = {}, p4 = {};
    const float* aRe = Are + (size_t)(row0 + m) * K;
    const float* aIm = Aim + (size_t)(row0 + m) * K;

    if (wave == 0) {
        tdm_load_tile((unsigned)(size_t)&sBre[0][0], Bre + col0, CH, Nn);
        tdm_load_tile((unsigned)(size_t)&sBim[0][0], Bim + col0, CH, Nn);
    }
    for (int c = 0; c < nch; ++c) {
        const int buf = c & 1;
        if (wave == 0) {
            if (c + 1 < nch) {   // prefetch next chunk into the other buffer
                const size_t gofs = (size_t)(c + 1) * CH * Nn + col0;
                tdm_load_tile((unsigned)(size_t)&sBre[buf ^ 1][0], Bre + gofs, CH, Nn);
                tdm_load_tile((unsigned)(size_t)&sBim[buf ^ 1][0], Bim + gofs, CH, Nn);
                __builtin_amdgcn_s_wait_tensorcnt(2);   // chunk c (2 older ops) complete
            } else {
                __builtin_amdgcn_s_wait_tensorcnt(0);
            }
        }
        __syncthreads();        // publish chunk c to all waves
        const int kbase = c * CH;
        __builtin_prefetch(aRe + kbase + CH, 0, 1);
        for (int kk = 0; kk < CH; kk += 4) {
            const int ka = kbase + kk + koff;
            v2f ar = { aRe[ka], aRe[ka + 1] };
            v2f ai = { aIm[ka], aIm[ka + 1] };
            const float* brp = &sBre[buf][(kk + koff) * 16 + m];
            const float* bip = &sBim[buf][(kk + koff) * 16 + m];
            v2f br = { brp[0], brp[16] };
            v2f bi = { bip[0], bip[16] };
            p1 = __builtin_amdgcn_wmma_f32_16x16x4_f32(false, ar, false, br, (short)0, p1, false, false);
            p2 = __builtin_amdgcn_wmma_f32_16x16x4_f32(false, ai, false, bi, (short)0, p2, false, false);
            p3 = __builtin_amdgcn_wmma_f32_16x16x4_f32(false, ar, false, bi, (short)0, p3, false, false);
            p4 = __builtin_amdgcn_wmma_f32_16x16x4_f32(false, ai, false, br, (short)0, p4, false, false);
        }
        __syncthreads();        // all reads of buf done before it is overwritten
    }

    const int col = col0 + m;
#pragma unroll
    for (int v = 0; v < 8; ++v) {
        const int row = row0 + v + hi * 8;          // C/D layout: hi lanes own rows 8..15
        const float yre = p1[v] - p2[v];
        const float yim = p3[v] + p4[v];
        const size_t o = (size_t)row * Nn + col;
        if (mode == 0) {        // twiddle e^{2pi i k1 t2/NFFT}; k1*t2 < 384000, no mod needed
            double s, c;
            sincos((double)((long long)row * col) * (2.0 * PI_D / (double)NFFT_), &s, &c);
            Cre[o] = yre * (float)c - yim * (float)s;
            Cim[o] = yre * (float)s + yim * (float)c;
        } else {
            Cre[o] = yre * (1.0f / (float)NFFT_);
        }
    }
}

// ---------------- max |h| reduction + normalize ----------------
__global__ void max_partial(const float* __restrict__ hu, float* __restrict__ part) {
    __shared__ float sm[256];
    const int tid = threadIdx.x;
    float v = 0.0f;
    for (int idx = blockIdx.x * 256 + tid; idx < NFFT_; idx += 256 * 256)
        v = fmaxf(v, fabsf(hu[idx]));
    sm[tid] = v;
    __syncthreads();
    for (int s = 128; s > 0; s >>= 1) {
        if (tid < s) sm[tid] = fmaxf(sm[tid], sm[tid + s]);
        __syncthreads();
    }
    if (tid == 0) part[blockIdx.x] = sm[0];
}

__global__ void max_final(float* __restrict__ part) {
    __shared__ float sm[256];
    const int tid = threadIdx.x;
    sm[tid] = part[tid];
    __syncthreads();
    for (int s = 128; s > 0; s >>= 1) {
        if (tid < s) sm[tid] = fmaxf(sm[tid], sm[tid + s]);
        __syncthreads();
    }
    if (tid == 0) part[256] = sm[0];
}

__global__ void scale_kernel(const float* __restrict__ hu, const float* __restrict__ maxv,
                             float* __restrict__ outh) {
    const int idx = blockIdx.x * blockDim.x + threadIdx.x;
    if (idx < NFFT_) outh[idx] = hu[idx] / maxv[0];
}

// ---------------- host launcher ----------------
extern "C" void kernel_launch(void* const* d_in, const int* in_sizes, int n_in,
                              void* d_out, int out_size, void* d_ws, size_t ws_size,
                              hipStream_t stream) {
    (void)in_sizes; (void)n_in; (void)out_size; (void)ws_size;
    const float* xr = (const float*)d_in[0];
    const float* xi = (const float*)d_in[1];
    const float* B  = (const float*)d_in[2];
    const float* C  = (const float*)d_in[3];
    const float* X  = (const float*)d_in[4];
    float* ws  = (float*)d_ws;
    float* out = (float*)d_out;
    float* outH = out;                                   // [F,16] complex64 interleaved
    float* outh = out + (size_t)FBINS * NM * 2;          // [NFFT] float32

    prep_kernel<<<1, 256, 0, stream>>>(X, ws);

    solve_kernel<<<(FBINS + 15) / 16, 256, 0, stream>>>(xr, xi, B, C, ws, outH, ws + OFF_HS);

    w_kernel<<<(NN2 * NN2 + 255) / 256, 256, 0, stream>>>(ws + OFF_W2RE, ws + OFF_W2IM, NN2);
    w_kernel<<<(NN1 * NN1 + 255) / 256, 256, 0, stream>>>(ws + OFF_W1RE, ws + OFF_W1IM, NN1);
    pack_kernel<<<(NFFT_ + 255) / 256, 256, 0, stream>>>(ws + OFF_HS, ws + OFF_XRE, ws + OFF_XIM);

    const int colTiles = NN2 / 16;                       // 50
    const int strips   = (NN1 + 127) / 128;              // 4
    const int blocks   = strips * colTiles;              // 200
    // Step A + B: Y = Xm * W800, twiddled -> Z
    gemm_cplx<<<blocks, 256, 0, stream>>>(ws + OFF_XRE, ws + OFF_XIM,
                                          ws + OFF_W2RE, ws + OFF_W2IM,
                                          ws + OFF_ZRE, ws + OFF_ZIM,
                                          NN1, NN2, NN2, 0);
    // Step C: h[t1*800+t2] = Re( W480 * Z ) / NFFT
    gemm_cplx<<<blocks, 256, 0, stream>>>(ws + OFF_W1RE, ws + OFF_W1IM,
                                          ws + OFF_ZRE, ws + OFF_ZIM,
                                          ws + OFF_HU, ws + OFF_ZIM,
                                          NN1, NN1, NN2, 1);

    max_partial<<<256, 256, 0, stream>>>(ws + OFF_HU, ws + OFF_MAXP);
    max_final<<<1, 256, 0, stream>>>(ws + OFF_MAXP);
    scale_kernel<<<(NFFT_ + 255) / 256, 256, 0, stream>>>(ws + OFF_HU, ws + OFF_MAXP + 256, outh);
}